// GNNWithAttention_44753559224619
// MI455X (gfx1250) — compile-verified
//
#include <hip/hip_runtime.h>

// ---------------------------------------------------------------------------
// 2-layer single-head GAT for gfx1250 (MI455X).
//   GEMMs      : V_WMMA_F32_16X16X4_F32, one 16x16 tile per wave32.
//   Softmax    : segment max via monotonic-u32 atomicMax, segment sum via
//                global f32 atomicAdd (L2-resident, 192MB L2 holds all data).
//   Aggregate  : 16 lanes/edge, float4 gathers of h[src], f32 atomic adds.
// ---------------------------------------------------------------------------

typedef float v2f __attribute__((ext_vector_type(2)));
typedef float v8f __attribute__((ext_vector_type(8)));

#define NEG_SLOPE 0.2f
#define EPSV 1e-16f
#define FDIM 64

// order-preserving f32 <-> u32 encoding (for atomicMax-based segment max)
__device__ __forceinline__ unsigned enc_f32(float f) {
  unsigned u = __float_as_uint(f);
  return (u & 0x80000000u) ? ~u : (u | 0x80000000u);
}
__device__ __forceinline__ float dec_f32(unsigned u) {
  u = (u & 0x80000000u) ? (u & 0x7FFFFFFFu) : ~u;
  return __uint_as_float(u);
}
#define ENC_NEG_INF 0x007FFFFFu  // enc_f32(-inf)

// ---------------------------------------------------------------------------
// H[M x 64] = X[M x K] @ W[K x 64] via V_WMMA_F32_16X16X4_F32.
// 128 threads = 4 waves; wave w computes the 16x16 tile at cols [16w,16w+16).
// A frag (16x4): lane l: row = l&15, holds K = (l>>4)*2 + {0,1}
// B frag (4x16): lane l: col = l&15, holds K = (l>>4)*2 + {0,1}
// C/D (16x16) : vgpr v, lane l: row = v + (l>>4)*8, col = l&15
// ---------------------------------------------------------------------------
template <int K>
__global__ __launch_bounds__(128) void gat_gemm_wmma(
    const float* __restrict__ X, const float* __restrict__ W,
    float* __restrict__ H, int M) {
  const int wave = threadIdx.x >> 5;
  const int lane = threadIdx.x & 31;
  const int l16  = lane & 15;
  const int kh   = (lane >> 4) << 1;  // 0 or 2
  const int rowBase = blockIdx.x * 16;
  const int colBase = wave * 16;

  int arow = rowBase + l16;
  if (arow >= M) arow = M - 1;  // M is a multiple of 16 here; safety clamp
  const float* xr = X + (size_t)arow * K;
  const float* wc = W + colBase + l16;

  v8f c = {0.f, 0.f, 0.f, 0.f, 0.f, 0.f, 0.f, 0.f};
#pragma unroll
  for (int k = 0; k < K; k += 4) {
    v2f a, b;
    a[0] = xr[k + kh];
    a[1] = xr[k + kh + 1];
    b[0] = wc[(size_t)(k + kh) * FDIM];
    b[1] = wc[(size_t)(k + kh + 1) * FDIM];
    c = __builtin_amdgcn_wmma_f32_16x16x4_f32(false, a, false, b,
                                              (short)0, c, false, false);
  }

  const int rOff = (lane >> 4) << 3;  // 0 or 8
#pragma unroll
  for (int v = 0; v < 8; ++v) {
    int r = rowBase + rOff + v;
    if (r >= M) r = M - 1;
    H[(size_t)r * FDIM + colBase + l16] = c[v];
  }
}

// per-node attention projections: hs = h . a_src, hd = h . a_dst
__global__ void gat_node_dots(const float* __restrict__ H,
                              const float* __restrict__ a_src,
                              const float* __restrict__ a_dst,
                              float* __restrict__ hs, float* __restrict__ hd,
                              int Nn) {
  int i = blockIdx.x * blockDim.x + threadIdx.x;
  if (i >= Nn) return;
  const float* h = H + (size_t)i * FDIM;
  float s = 0.f, d = 0.f;
#pragma unroll
  for (int f = 0; f < FDIM; ++f) {
    float v = h[f];
    s += v * a_src[f];
    d += v * a_dst[f];
  }
  hs[i] = s;
  hd[i] = d;
}

// zero the aggregation buffer; reset per-node max/denom
__global__ void gat_init(float* __restrict__ agg, unsigned* __restrict__ maxenc,
                         float* __restrict__ denom, int Nn) {
  int i = blockIdx.x * blockDim.x + threadIdx.x;
  if (i < Nn * FDIM) agg[i] = 0.f;
  if (i < Nn) {
    maxenc[i] = ENC_NEG_INF;
    denom[i]  = 0.f;
  }
}

// e = leaky_relu(hs[src] + hd[dst]); segment max via encoded atomicMax
__global__ void gat_edge_logits(const int* __restrict__ ei,
                                const float* __restrict__ hs,
                                const float* __restrict__ hd,
                                float* __restrict__ ebuf,
                                unsigned* __restrict__ maxenc, int E, int ET) {
  int e = blockIdx.x * blockDim.x + threadIdx.x;
  if (e >= ET) return;
  int s, d;
  if (e < E) { s = ei[e]; d = ei[E + e]; } else { s = d = e - E; }
  float v = hs[s] + hd[d];
  v = (v > 0.f) ? v : NEG_SLOPE * v;
  ebuf[e] = v;
  atomicMax(&maxenc[d], enc_f32(v));
}

// ex = exp(e - m[dst]); segment sum of ex into denom
__global__ void gat_edge_exp(const int* __restrict__ ei,
                             float* __restrict__ ebuf,
                             const unsigned* __restrict__ maxenc,
                             float* __restrict__ denom, int E, int ET) {
  int e = blockIdx.x * blockDim.x + threadIdx.x;
  if (e >= ET) return;
  int d = (e < E) ? ei[E + e] : (e - E);
  float ex = __expf(ebuf[e] - dec_f32(maxenc[d]));
  ebuf[e] = ex;
  atomicAdd(&denom[d], ex);
}

// alpha = ex/(denom+eps); agg[dst] += alpha * h[src]  (16 lanes x float4/edge)
__global__ void gat_edge_agg(const int* __restrict__ ei,
                             const float* __restrict__ ebuf,
                             const float* __restrict__ denom,
                             const float* __restrict__ H,
                             float* __restrict__ agg,
                             float* __restrict__ alpha_out, int E, int ET) {
  long long gid = (long long)blockIdx.x * blockDim.x + threadIdx.x;
  int e   = (int)(gid >> 4);
  int sub = (int)(gid & 15);
  if (e >= ET) return;
  int s, d;
  if (e < E) { s = ei[e]; d = ei[E + e]; } else { s = d = e - E; }
  float alpha = ebuf[e] / (denom[d] + EPSV);
  if (alpha_out != nullptr && sub == 0) alpha_out[e] = alpha;
  const float4 hv = ((const float4*)(H + (size_t)s * FDIM))[sub];
  float* ad = agg + (size_t)d * FDIM + sub * 4;
  atomicAdd(ad + 0, alpha * hv.x);
  atomicAdd(ad + 1, alpha * hv.y);
  atomicAdd(ad + 2, alpha * hv.z);
  atomicAdd(ad + 3, alpha * hv.w);
}

// out = relu(agg + b)
__global__ void gat_finalize(float* __restrict__ agg, const float* __restrict__ b,
                             int total) {
  int i = blockIdx.x * blockDim.x + threadIdx.x;
  if (i >= total) return;
  float v = agg[i] + b[i & (FDIM - 1)];
  agg[i] = (v > 0.f) ? v : 0.f;
}

static inline int cdiv_ll(long long a, long long b) {
  return (int)((a + b - 1) / b);
}

extern "C" void kernel_launch(void* const* d_in, const int* in_sizes, int n_in,
                              void* d_out, int out_size, void* d_ws,
                              size_t ws_size, hipStream_t stream) {
  const float* x      = (const float*)d_in[0];
  const int*   ei     = (const int*)d_in[1];
  const float* W1     = (const float*)d_in[2];
  const float* a_src1 = (const float*)d_in[3];
  const float* a_dst1 = (const float*)d_in[4];
  const float* b1     = (const float*)d_in[5];
  const float* W2     = (const float*)d_in[6];
  const float* a_src2 = (const float*)d_in[7];
  const float* a_dst2 = (const float*)d_in[8];
  const float* b2     = (const float*)d_in[9];

  const int IN_DIM = 128;
  const int Nn = in_sizes[0] / IN_DIM;
  const int E  = in_sizes[1] / 2;
  const int ET = E + Nn;  // edges + self loops (self loops appended, id = E+i)

  // workspace layout (floats)
  float*    bufH   = (float*)d_ws;                  // Nn*64 transformed feats
  float*    bufAgg = bufH + (size_t)Nn * FDIM;      // Nn*64 layer-1 output h1
  float*    hs     = bufAgg + (size_t)Nn * FDIM;    // Nn
  float*    hd     = hs + Nn;                       // Nn
  unsigned* maxenc = (unsigned*)(hd + Nn);          // Nn
  float*    denom  = (float*)(maxenc + Nn);         // Nn
  float*    ebuf   = denom + Nn;                    // ET (e -> ex in place)

  float* h2out     = (float*)d_out;                 // Nn*64
  float* alpha_out = h2out + (size_t)Nn * FDIM;     // ET

  const int gemmGrid = cdiv_ll(Nn, 16);
  const int nodeGrid = cdiv_ll(Nn, 256);
  const int nfGrid   = cdiv_ll((long long)Nn * FDIM, 256);
  const int edgeGrid = cdiv_ll(ET, 256);
  const int aggGrid  = cdiv_ll((long long)ET * 16, 256);

  // ---------------- layer 1 ----------------
  gat_gemm_wmma<128><<<gemmGrid, 128, 0, stream>>>(x, W1, bufH, Nn);
  gat_node_dots<<<nodeGrid, 256, 0, stream>>>(bufH, a_src1, a_dst1, hs, hd, Nn);
  gat_init<<<nfGrid, 256, 0, stream>>>(bufAgg, maxenc, denom, Nn);
  gat_edge_logits<<<edgeGrid, 256, 0, stream>>>(ei, hs, hd, ebuf, maxenc, E, ET);
  gat_edge_exp<<<edgeGrid, 256, 0, stream>>>(ei, ebuf, maxenc, denom, E, ET);
  gat_edge_agg<<<aggGrid, 256, 0, stream>>>(ei, ebuf, denom, bufH, bufAgg,
                                            nullptr, E, ET);
  gat_finalize<<<nfGrid, 256, 0, stream>>>(bufAgg, b1, Nn * FDIM);  // -> h1

  // ---------------- layer 2 ----------------
  gat_gemm_wmma<64><<<gemmGrid, 128, 0, stream>>>(bufAgg, W2, bufH, Nn);
  gat_node_dots<<<nodeGrid, 256, 0, stream>>>(bufH, a_src2, a_dst2, hs, hd, Nn);
  gat_init<<<nfGrid, 256, 0, stream>>>(h2out, maxenc, denom, Nn);
  gat_edge_logits<<<edgeGrid, 256, 0, stream>>>(ei, hs, hd, ebuf, maxenc, E, ET);
  gat_edge_exp<<<edgeGrid, 256, 0, stream>>>(ei, ebuf, maxenc, denom, E, ET);
  gat_edge_agg<<<aggGrid, 256, 0, stream>>>(ei, ebuf, denom, bufH, h2out,
                                            alpha_out, E, ET);
  gat_finalize<<<nfGrid, 256, 0, stream>>>(h2out, b2, Nn * FDIM);  // relu(h2)
}